// NSSM_54520314855991
// MI455X (gfx1250) — compile-verified
//
#include <hip/hip_runtime.h>
#include <hip/hip_bf16.h>
#include <math.h>

// Problem sizes (fixed by the reference)
#define S 16      // states
#define C 8       // chains
#define N 5000    // nodes (divisible by 4 -> whole-quad b128 stores)
#define T 100     // iterations

typedef __attribute__((ext_vector_type(2))) float v2f;
typedef __attribute__((ext_vector_type(4))) float v4f;
typedef __attribute__((ext_vector_type(8))) float v8f;

// ---------------------------------------------------------------------------
// K1: single wave32. log_softmax(state_init), log_softmax rows(transition),
// then the 99-step HMM recursion as exp-domain vector-matrix products via
// V_WMMA_F32_16X16X4_F32 (A = p replicated into all 16 rows, B = exp(logT);
// 4 chained K=4 WMMAs accumulate the full K=16 contraction; every row of the
// 16x16 D equals p.P, so lane L reads its own acc[0] = p_new[L%16]).
// Writes log_hidden_state_probs to d_out region 3 and to workspace for K3.
// ---------------------------------------------------------------------------
__global__ __launch_bounds__(32) void k1_recursion(
    const float* __restrict__ usi,   // (16)
    const float* __restrict__ utm,   // (16,16)
    float* __restrict__ hs_ws,       // (T,16) workspace
    float* __restrict__ out2)        // (T,16) output region 3
{
  __shared__ float p_lds[16];        // current state marginal, prob domain
  __shared__ float tmp[16];
  __shared__ float pexp_s[16][16];   // exp(log_softmax(transition)) = row-stochastic P

  const int L  = threadIdx.x;
  const int lo = L & 15;
  const int hi = L >> 4;             // 0 or 1 (lane half)

  // ---- log_softmax of state init ----
  if (L < 16) tmp[L] = usi[L];
  __syncthreads();
  float m = -INFINITY;
  for (int i = 0; i < 16; ++i) m = fmaxf(m, tmp[i]);
  float ssum = 0.f;
  for (int i = 0; i < 16; ++i) ssum += __expf(tmp[i] - m);
  const float lse0 = m + __logf(ssum);
  const float lsi  = tmp[lo] - lse0;
  if (L < 16) {
    out2[L]  = lsi;
    hs_ws[L] = lsi;
    p_lds[L] = __expf(lsi);
  }

  // ---- per-row log_softmax of transition; store exp() (row-stochastic P) ----
  if (L < 16) {
    float rm = -INFINITY;
    for (int j = 0; j < 16; ++j) rm = fmaxf(rm, utm[L * 16 + j]);
    float rs = 0.f;
    for (int j = 0; j < 16; ++j) rs += __expf(utm[L * 16 + j] - rm);
    const float rlse = rm + __logf(rs);
    for (int j = 0; j < 16; ++j) pexp_s[L][j] = __expf(utm[L * 16 + j] - rlse);
  }
  __syncthreads();

  // ---- preload B operands (4 K-chunks of 4). 32-bit operand layout:
  // element (K,N): VGPR v holds K = 4*chunk + v + 2*hi, N = lane%16.
  v2f b0, b1, b2, b3;
  b0.x = pexp_s[0  + 2*hi][lo]; b0.y = pexp_s[1  + 2*hi][lo];
  b1.x = pexp_s[4  + 2*hi][lo]; b1.y = pexp_s[5  + 2*hi][lo];
  b2.x = pexp_s[8  + 2*hi][lo]; b2.y = pexp_s[9  + 2*hi][lo];
  b3.x = pexp_s[12 + 2*hi][lo]; b3.y = pexp_s[13 + 2*hi][lo];

  for (int t = 1; t < T; ++t) {
    // A: every row M carries p (rows identical) -> lane value independent of M
    v2f a0, a1, a2, a3;
    a0.x = p_lds[0  + 2*hi]; a0.y = p_lds[1  + 2*hi];
    a1.x = p_lds[4  + 2*hi]; a1.y = p_lds[5  + 2*hi];
    a2.x = p_lds[8  + 2*hi]; a2.y = p_lds[9  + 2*hi];
    a3.x = p_lds[12 + 2*hi]; a3.y = p_lds[13 + 2*hi];

    v8f acc = {};
    acc = __builtin_amdgcn_wmma_f32_16x16x4_f32(false, a0, false, b0, (short)0, acc, false, false);
    acc = __builtin_amdgcn_wmma_f32_16x16x4_f32(false, a1, false, b1, (short)0, acc, false, false);
    acc = __builtin_amdgcn_wmma_f32_16x16x4_f32(false, a2, false, b2, (short)0, acc, false, false);
    acc = __builtin_amdgcn_wmma_f32_16x16x4_f32(false, a3, false, b3, (short)0, acc, false, false);

    const float pn = acc[0];      // D row 0 (==all rows): p_new[lane%16]
    const float hn = __logf(pn);

    __syncthreads();
    if (L < 16) {
      p_lds[L]          = pn;
      hs_ws[t * 16 + L] = hn;
      out2[t * 16 + L]  = hn;
    }
    __syncthreads();
  }
}

// ---------------------------------------------------------------------------
// K2: per-state logsumexp over the N=5000 emission row (the axis=1 softmax
// denominator). 16 blocks x 256 threads, classic two-phase LDS reduction.
// ---------------------------------------------------------------------------
__global__ __launch_bounds__(256) void k2_row_lse(
    const float* __restrict__ em,    // (16,N)
    float* __restrict__ lse_ws)      // (16)
{
  const int s = blockIdx.x;
  const float* row = em + (size_t)s * N;
  __shared__ float red[256];
  const int tid = threadIdx.x;

  float m = -INFINITY;
  for (int i = tid; i < N; i += 256) m = fmaxf(m, row[i]);
  red[tid] = m; __syncthreads();
  for (int off = 128; off > 0; off >>= 1) {
    if (tid < off) red[tid] = fmaxf(red[tid], red[tid + off]);
    __syncthreads();
  }
  m = red[0]; __syncthreads();

  float ssum = 0.f;
  for (int i = tid; i < N; i += 256) ssum += __expf(row[i] - m);
  red[tid] = ssum; __syncthreads();
  for (int off = 128; off > 0; off >>= 1) {
    if (tid < off) red[tid] += red[tid + off];
    __syncthreads();
  }
  if (tid == 0) lse_ws[s] = m + __logf(red[0]);
}

// ---------------------------------------------------------------------------
// K3: the bandwidth-bound streamer. One block = (one 64-node tile, one t).
// 258 MB of stores total -> ~11 us floor at 23.3 TB/s, so everything is
// organized around wide streaming stores:
//   thread = (row 0..15, quad 0..15); lane's chain c = (tid>>4)&7 is CONSTANT,
//   state s = (tid>>7) + 2k across the k=0..7 row loop. Per row:
//   one ds_load_b128 (emission quad), one packed add, one nontemporal
//   global_store_b128 (wave = 2 rows x 16 lanes x 16B = two 256B bursts).
// ---------------------------------------------------------------------------
#define NT 64
__global__ __launch_bounds__(256) void k3_stream(
    const float* __restrict__ em,     // (16,N)
    const float* __restrict__ cw,     // (N,8) (leading dim-1 axis dropped)
    const float* __restrict__ hs_ws,  // (T,16)
    const float* __restrict__ lse_ws, // (16)
    float* __restrict__ out0,         // (T,N)
    float* __restrict__ out1)         // (T,16,8,N)
{
  const int n0  = blockIdx.x * NT;
  const int t   = blockIdx.y;
  const int tid = threadIdx.x;

  __shared__ __align__(16) float le_s[S][NT];    // normalized log_emission tile
  __shared__ __align__(16) float lcwT[C][NT];    // normalized log_chain_weights^T
  __shared__ float hsh[S];

  if (tid < S) hsh[tid] = hs_ws[t * S + tid];

  const int q   = tid & 15;        // quad index: nodes nq0..nq0+3
  const int nq0 = q * 4;
  const int rb  = tid >> 4;        // 0..15
  // whole-quad validity: N % 4 == 0, so quads are never partial
  const bool qvalid = (n0 + nq0 + 3) < N;

  // ---- load + normalize emission tile: 16 rows x 16 quads, one v4f/thread
  {
    const int s = tid >> 4;        // 0..15
    v4f le4 = {};
    if (qvalid) {
      le4 = *(const v4f*)(em + (size_t)s * N + (n0 + nq0));
      le4 = le4 - lse_ws[s];
    }
    *(v4f*)&le_s[s][nq0] = le4;
  }

  // ---- chain-weight log_softmax for this tile's nodes (thread per node)
  if (tid < NT) {
    const int n = n0 + tid;
    if (n < N) {
      const v4f xa = *(const v4f*)(cw + (size_t)n * C);
      const v4f xb = *(const v4f*)(cw + (size_t)n * C + 4);
      float x[C] = {xa.x, xa.y, xa.z, xa.w, xb.x, xb.y, xb.z, xb.w};
      float mm = x[0];
      for (int c = 1; c < C; ++c) mm = fmaxf(mm, x[c]);
      float ssum = 0.f;
      for (int c = 0; c < C; ++c) ssum += __expf(x[c] - mm);
      const float lse = mm + __logf(ssum);
      for (int c = 0; c < C; ++c) lcwT[c][tid] = x[c] - lse;
    }
  }
  __syncthreads();

  // ---- streaming phase: this thread's chain is fixed, states stride by 2
  const int c  = rb & 7;           // constant chain for this thread
  const int s0 = rb >> 3;          // state parity (0 or 1)

  const v4f lq = *(const v4f*)&lcwT[c][nq0];   // chain-weight quad (b128)

  const size_t tbase = (size_t)t * S * C * N + (size_t)(n0 + nq0);
  #pragma unroll
  for (int k = 0; k < 8; ++k) {
    const int s = s0 + 2 * k;
    const v4f le4 = *(const v4f*)&le_s[s][nq0];          // ds_load_b128
    const v4f val = le4 + (lq + hsh[s]);                 // packed adds
    if (qvalid) {
      float* p = out1 + tbase + (size_t)(s * C + c) * N; // 16B aligned
      __builtin_nontemporal_store(val, (v4f*)p);         // global_store_b128 nt
    }
  }

  // ---- out0[t,n] = logsumexp_s(h[t,s] + log_em[s,n]) (logsumexp_c(lcw)==0)
  if (tid < NT) {
    const int n = n0 + tid;
    if (n < N) {
      float mm = -INFINITY;
      for (int s = 0; s < S; ++s) mm = fmaxf(mm, hsh[s] + le_s[s][tid]);
      float ssum = 0.f;
      for (int s = 0; s < S; ++s) ssum += __expf(hsh[s] + le_s[s][tid] - mm);
      __builtin_nontemporal_store(mm + __logf(ssum), &out0[(size_t)t * N + n]);
    }
  }
}

// ---------------------------------------------------------------------------
extern "C" void kernel_launch(void* const* d_in, const int* in_sizes, int n_in,
                              void* d_out, int out_size, void* d_ws, size_t ws_size,
                              hipStream_t stream) {
  const float* usi = (const float*)d_in[0];   // (16)
  const float* ucw = (const float*)d_in[1];   // (1,N,8)
  const float* uem = (const float*)d_in[2];   // (16,N)
  const float* utm = (const float*)d_in[3];   // (16,16)

  float* out0 = (float*)d_out;                          // (T,N)
  float* out1 = out0 + (size_t)T * N;                   // (T,16,8,N)
  float* out2 = out1 + (size_t)T * S * C * N;           // (T,16)

  float* hs_ws  = (float*)d_ws;                         // T*16 floats
  float* lse_ws = hs_ws + (size_t)T * S;                // 16 floats

  k1_recursion<<<1, 32, 0, stream>>>(usi, utm, hs_ws, out2);
  k2_row_lse<<<S, 256, 0, stream>>>(uem, lse_ws);

  dim3 grid((N + NT - 1) / NT, T);                      // 79 x 100 blocks
  k3_stream<<<grid, 256, 0, stream>>>(uem, ucw, hs_ws, lse_ws, out0, out1);
}